// Performer_66185446031841
// MI455X (gfx1250) — compile-verified
//
#include <hip/hip_runtime.h>
#include <math.h>

#define BH    128
#define NSEQ  4096
#define DDIM  64
#define MFEAT 256

// data_normalizer = d^-0.25 = 64^-0.25
#define DN     0.35355339059327373f
// ratio = m^-0.5 = 256^-0.5
#define RATIO  0.0625f
#define FEPS   1e-4f

typedef float v2f __attribute__((ext_vector_type(2)));
typedef float v8f __attribute__((ext_vector_type(8)));

// D = A(16x4 f32) * B(4x16 f32) + C(16x16 f32)  -- CDNA5 fp32 WMMA
__device__ __forceinline__ v8f wmma4(v2f a, v2f b, v8f c) {
  return __builtin_amdgcn_wmma_f32_16x16x4_f32(false, a, false, b, (short)0, c,
                                               false, false);
}

__device__ __forceinline__ void atomicMaxF(float* addr, float val) {
  unsigned int* ai = (unsigned int*)addr;
  unsigned int old = *ai;
  while (__uint_as_float(old) < val) {
    unsigned int assumed = old;
    old = atomicCAS(ai, assumed, __float_as_uint(val));
    if (old == assumed) break;
  }
}

// ---------------------------------------------------------------- init
__global__ void init_ws(float* ctx, float* ksum, float* kstab) {
  int idx = blockIdx.x * blockDim.x + threadIdx.x;
  if (idx < BH * MFEAT * DDIM) ctx[idx] = 0.f;
  if (idx < BH * MFEAT)        ksum[idx] = 0.f;
  if (idx < BH)                kstab[idx] = -3.0e38f;
}

// ---------------------------------------------------------------- pass 1
// Per block: one (bh, 512-row chunk). k_dash via WMMA, global max -> kstab[bh].
// proj B-fragments are loop-invariant -> hoisted to registers (64 VGPRs).
__global__ void kmax_kernel(const float* __restrict__ K,
                            const float* __restrict__ P,
                            float* __restrict__ kstab) {
  extern __shared__ float smem[];
  float* s_proj = smem;                 // [256][64]
  float* s_k    = smem + 16384;         // [16][64] (prescaled by DN)
  float* s_red  = smem + 16384 + 1024;  // [256]
  const int tid  = threadIdx.x;
  const int w    = tid >> 5;
  const int lane = tid & 31;
  const int l15  = lane & 15;
  const int hi2  = (lane >> 4) << 1;    // K-pair offset per ISA A/B layout
  const int bh    = blockIdx.x >> 3;
  const int chunk = blockIdx.x & 7;

  for (int i = tid; i < MFEAT * DDIM; i += 256) s_proj[i] = P[i];
  __syncthreads();

  // hoist proj fragments: wave covers feature-col tiles {2w, 2w+1}
  v2f bfrag[2][16];
#pragma unroll
  for (int t = 0; t < 2; ++t) {
    int j0 = ((w << 1) + t) << 4;
#pragma unroll
    for (int ks = 0; ks < 16; ++ks) {
      int kk = (ks << 2) + hi2;
      bfrag[t][ks][0] = s_proj[(j0 + l15) * DDIM + kk];
      bfrag[t][ks][1] = s_proj[(j0 + l15) * DDIM + kk + 1];
    }
  }

  const float* kbase = K + ((size_t)bh * NSEQ + (size_t)chunk * 512) * DDIM;
  float runmax = -3.0e38f;
  for (int step = 0; step < 32; ++step) {
    __syncthreads();
    for (int i = tid; i < 1024; i += 256) s_k[i] = DN * kbase[step * 1024 + i];
    __syncthreads();
    v8f acc0 = {}, acc1 = {};
#pragma unroll
    for (int ks = 0; ks < 16; ++ks) {   // K = 64, 4 per wmma; A shared by tiles
      int kk = (ks << 2) + hi2;
      v2f a;
      a[0] = s_k[l15 * DDIM + kk];
      a[1] = s_k[l15 * DDIM + kk + 1];
      acc0 = wmma4(a, bfrag[0][ks], acc0);
      acc1 = wmma4(a, bfrag[1][ks], acc1);
    }
#pragma unroll
    for (int v = 0; v < 8; ++v)
      runmax = fmaxf(runmax, fmaxf(acc0[v], acc1[v]));
  }
  __syncthreads();
  s_red[tid] = runmax;
  __syncthreads();
  for (int s = 128; s > 0; s >>= 1) {
    if (tid < s) s_red[tid] = fmaxf(s_red[tid], s_red[tid + s]);
    __syncthreads();
  }
  if (tid == 0) atomicMaxF(&kstab[bh], s_red[0]);
}

// ---------------------------------------------------------------- pass 2
// Per block: (bh, 512-row chunk). k' in LDS; context += k'^T v via WMMA
// (per-wave register tiles), ksum via column sums; atomicAdd epilogue.
__global__ void kfeat_kernel(const float* __restrict__ K,
                             const float* __restrict__ V,
                             const float* __restrict__ P,
                             const float* __restrict__ kstab,
                             float* __restrict__ ctx,
                             float* __restrict__ ksum) {
  extern __shared__ float smem[];
  float* s_proj = smem;            // 16384
  float* s_k    = s_proj + 16384;  // 1024 (prescaled)
  float* s_v    = s_k + 1024;      // 1024
  float* s_kp   = s_v + 1024;      // 4096  [16 n][256 m]
  float* s_diag = s_kp + 4096;     // 16
  const int tid  = threadIdx.x;
  const int w    = tid >> 5;
  const int lane = tid & 31;
  const int l15  = lane & 15;
  const int hi2  = (lane >> 4) << 1;
  const int bh    = blockIdx.x >> 3;
  const int chunk = blockIdx.x & 7;

  for (int i = tid; i < MFEAT * DDIM; i += 256) s_proj[i] = P[i];
  __syncthreads();

  // hoist loop-invariant proj B-fragments for the dash GEMM
  v2f bfrag[2][16];
#pragma unroll
  for (int t = 0; t < 2; ++t) {
    int j0 = ((w << 1) + t) << 4;
#pragma unroll
    for (int ks = 0; ks < 16; ++ks) {
      int kk = (ks << 2) + hi2;
      bfrag[t][ks][0] = s_proj[(j0 + l15) * DDIM + kk];
      bfrag[t][ks][1] = s_proj[(j0 + l15) * DDIM + kk + 1];
    }
  }

  const float stab = kstab[bh];
  const float* kbase = K + ((size_t)bh * NSEQ + (size_t)chunk * 512) * DDIM;
  const float* vbase = V + ((size_t)bh * NSEQ + (size_t)chunk * 512) * DDIM;

  v8f acc[2][4] = {};       // wave owns m-tiles {2w,2w+1} x e-tiles {0..3}
  float ksum_acc = 0.f;     // thread owns feature column `tid`

  for (int step = 0; step < 32; ++step) {
    __syncthreads();
    for (int i = tid; i < 1024; i += 256) {
      s_k[i] = DN * kbase[step * 1024 + i];
      s_v[i] = vbase[step * 1024 + i];
    }
    __syncthreads();
    if (tid < 16) {  // diag = 0.5 * dn^2 * ||k_row||^2 ; s_k already scaled
      float d = 0.f;
      for (int dd = 0; dd < DDIM; ++dd) { float x = s_k[tid * DDIM + dd]; d += x * x; }
      s_diag[tid] = 0.5f * d;
    }
    __syncthreads();
    // k_dash -> k' for this 16-row strip (wave: 2 feature-col tiles, shared A)
    {
      v8f d0 = {}, d1 = {};
#pragma unroll
      for (int ks = 0; ks < 16; ++ks) {
        int kk = (ks << 2) + hi2;
        v2f a;
        a[0] = s_k[l15 * DDIM + kk];
        a[1] = s_k[l15 * DDIM + kk + 1];
        d0 = wmma4(a, bfrag[0][ks], d0);
        d1 = wmma4(a, bfrag[1][ks], d1);
      }
      int j0 = (w << 1) << 4, j1 = ((w << 1) + 1) << 4;
#pragma unroll
      for (int v = 0; v < 8; ++v) {
        int r = v + ((lane >> 4) << 3);  // C layout: M = v (+8 for hi half)
        float ds = s_diag[r] + stab;
        s_kp[r * MFEAT + j0 + l15] = RATIO * (expf(d0[v] - ds) + FEPS);
        s_kp[r * MFEAT + j1 + l15] = RATIO * (expf(d1[v] - ds) + FEPS);
      }
    }
    __syncthreads();
#pragma unroll
    for (int i = 0; i < 16; ++i) ksum_acc += s_kp[i * MFEAT + tid];
    // context += k'^T (m x 16) @ v (16 x e)  -- WMMA, K = 16 rows
#pragma unroll
    for (int mt = 0; mt < 2; ++mt) {
      int m0 = ((w << 1) + mt) << 4;
#pragma unroll
      for (int et = 0; et < 4; ++et) {
        int e0 = et << 4;
        v8f c = acc[mt][et];
#pragma unroll
        for (int ks = 0; ks < 4; ++ks) {
          int kk = (ks << 2) + hi2;
          v2f a, b;
          a[0] = s_kp[kk * MFEAT + m0 + l15];        // A = kp^T
          a[1] = s_kp[(kk + 1) * MFEAT + m0 + l15];
          b[0] = s_v[kk * DDIM + e0 + l15];
          b[1] = s_v[(kk + 1) * DDIM + e0 + l15];
          c = wmma4(a, b, c);
        }
        acc[mt][et] = c;
      }
    }
  }
  atomicAdd(&ksum[bh * MFEAT + tid], ksum_acc);
  for (int mt = 0; mt < 2; ++mt) {
    int m0 = ((w << 1) + mt) << 4;
    for (int et = 0; et < 4; ++et) {
      int e0 = et << 4;
#pragma unroll
      for (int v = 0; v < 8; ++v) {
        int m = m0 + v + ((lane >> 4) << 3);
        atomicAdd(&ctx[((size_t)bh * MFEAT + m) * DDIM + e0 + l15], acc[mt][et][v]);
      }
    }
  }
}

// ---------------------------------------------------------------- pass 3
// Per block: (bh, 32-row chunk). q' in LDS; out = (q'@ctx) * 1/(q'@ksum).
__global__ void qout_kernel(const float* __restrict__ Q,
                            const float* __restrict__ P,
                            const float* __restrict__ ctx,
                            const float* __restrict__ ksum,
                            float* __restrict__ out) {
  extern __shared__ float smem[];
  float* s_proj = smem;             // 16384
  float* s_ctx  = s_proj + 16384;   // 16384  [256 m][64 e]
  float* s_ksum = s_ctx + 16384;    // 256
  float* s_q    = s_ksum + 256;     // 2048   [32][64] (prescaled)
  float* s_qp   = s_q + 2048;       // 8192   [32][256]
  float* s_diag = s_qp + 8192;      // 32
  float* s_stab = s_diag + 32;      // 32
  float* s_dinv = s_stab + 32;      // 32
  const int tid  = threadIdx.x;
  const int w    = tid >> 5;
  const int lane = tid & 31;
  const int l15  = lane & 15;
  const int hi2  = (lane >> 4) << 1;
  const int bh    = blockIdx.x >> 7;
  const int chunk = blockIdx.x & 127;
  const int n0    = chunk * 32;

  for (int i = tid; i < MFEAT * DDIM; i += 256) {
    s_proj[i] = P[i];
    s_ctx[i]  = ctx[(size_t)bh * MFEAT * DDIM + i];   // hits L2 (8 MB resident)
  }
  if (tid < MFEAT) s_ksum[tid] = ksum[bh * MFEAT + tid];
  const float* qbase = Q + ((size_t)bh * NSEQ + n0) * DDIM;
  for (int i = tid; i < 2048; i += 256) s_q[i] = DN * qbase[i];
  __syncthreads();
  if (tid < 32) {
    float d = 0.f;
    for (int dd = 0; dd < DDIM; ++dd) { float x = s_q[tid * DDIM + dd]; d += x * x; }
    s_diag[tid] = 0.5f * d;
  }
  __syncthreads();
  // q_dash: 2 row-tiles x 16 col-tiles = 32 tiles; each wave does 4
  for (int t = 0; t < 4; ++t) {
    int tile = w * 4 + t;
    int r0 = (tile >> 4) << 4, j0 = (tile & 15) << 4;
    v8f dacc = {};
#pragma unroll
    for (int ks = 0; ks < 16; ++ks) {
      int kk = (ks << 2) + hi2;
      v2f a, b;
      a[0] = s_q[(r0 + l15) * DDIM + kk];
      a[1] = s_q[(r0 + l15) * DDIM + kk + 1];
      b[0] = s_proj[(j0 + l15) * DDIM + kk];
      b[1] = s_proj[(j0 + l15) * DDIM + kk + 1];
      dacc = wmma4(a, b, dacc);
    }
#pragma unroll
    for (int v = 0; v < 8; ++v) {
      int r = r0 + v + ((lane >> 4) << 3);
      s_qp[r * MFEAT + j0 + l15] = dacc[v];
    }
  }
  __syncthreads();
  if (tid < 32) {                       // per-row stabilizer (is_query=True)
    float mx = -3.0e38f;
    for (int j = 0; j < MFEAT; ++j) mx = fmaxf(mx, s_qp[tid * MFEAT + j]);
    s_stab[tid] = mx;
  }
  __syncthreads();
  for (int i = tid; i < 32 * MFEAT; i += 256) {
    int r = i >> 8;
    s_qp[i] = RATIO * (expf(s_qp[i] - s_diag[r] - s_stab[r]) + FEPS);
  }
  __syncthreads();
  if (tid < 32) {
    float den = 0.f;
    for (int j = 0; j < MFEAT; ++j) den += s_qp[tid * MFEAT + j] * s_ksum[j];
    s_dinv[tid] = 1.f / den;
  }
  __syncthreads();
  // out tile per wave: (row-tile, e-tile) = (w>>2, w&3), K = 256
  {
    int r0 = (w >> 2) << 4, e0 = (w & 3) << 4;
    v8f oacc = {};
    for (int ks = 0; ks < 64; ++ks) {
      int kk = (ks << 2) + hi2;
      v2f a, b;
      a[0] = s_qp[(r0 + l15) * MFEAT + kk];
      a[1] = s_qp[(r0 + l15) * MFEAT + kk + 1];
      b[0] = s_ctx[kk * DDIM + e0 + l15];
      b[1] = s_ctx[(kk + 1) * DDIM + e0 + l15];
      oacc = wmma4(a, b, oacc);
    }
#pragma unroll
    for (int v = 0; v < 8; ++v) {
      int r = r0 + v + ((lane >> 4) << 3);
      out[((size_t)bh * NSEQ + n0 + r) * DDIM + e0 + l15] = oacc[v] * s_dinv[r];
    }
  }
}

// ---------------------------------------------------------------- launch
extern "C" void kernel_launch(void* const* d_in, const int* in_sizes, int n_in,
                              void* d_out, int out_size, void* d_ws, size_t ws_size,
                              hipStream_t stream) {
  const float* q = (const float*)d_in[0];
  const float* k = (const float*)d_in[1];
  const float* v = (const float*)d_in[2];
  const float* p = (const float*)d_in[3];
  float* ctx   = (float*)d_ws;                  // [128][256][64]
  float* ksum  = ctx + BH * MFEAT * DDIM;       // [128][256]
  float* kstab = ksum + BH * MFEAT;             // [128]
  float* out   = (float*)d_out;

  int initN = BH * MFEAT * DDIM;
  init_ws<<<(initN + 255) / 256, 256, 0, stream>>>(ctx, ksum, kstab);

  size_t sm1 = (16384 + 1024 + 256) * sizeof(float);
  kmax_kernel<<<BH * 8, 256, sm1, stream>>>(k, p, kstab);

  size_t sm2 = (16384 + 1024 + 1024 + 4096 + 16) * sizeof(float);
  kfeat_kernel<<<BH * 8, 256, sm2, stream>>>(k, v, p, kstab, ctx, ksum);

  size_t sm3 = (16384 + 16384 + 256 + 2048 + 8192 + 96) * sizeof(float);
  qout_kernel<<<BH * 128, 256, sm3, stream>>>(q, p, ctx, ksum, out);
}